// MMDLoss_19937238188450
// MI455X (gfx1250) — compile-verified
//
#include <hip/hip_runtime.h>
#include <hip/hip_bf16.h>
#include <math.h>

typedef float v2f __attribute__((ext_vector_type(2)));
typedef float v8f __attribute__((ext_vector_type(8)));

#define N_SAMPLES   4096
#define DIM         512
#define NUM_GROUPS  2
#define NUM_CLASSES 10
#define CG          (NUM_CLASSES * NUM_GROUPS)
#define EPS_F       1e-12f
#define W_M_F       1.0f
#define SIGMA_BASE_F 1.0f

// ---- workspace layout (float offsets) ----
#define OFF_NT        0            // [4096] teacher row sq-norms
#define OFF_NS        4096         // [4096] student row sq-norms
#define OFF_SUMT      8192         // [512]  column sums teacher
#define OFF_SUMS      8704         // [512]  column sums student
#define OFF_SUMNT     9216         // scalar sum of teacher norms
#define OFF_SUMNS     9217         // scalar sum of student norms
#define OFF_INVSCALE  9218         // scalar 1/(2*sigma_base*sigma_avg)
#define OFF_ACC_TT    9220         // [10]
#define OFF_ACC_SS    9230         // [20]
#define OFF_ACC_TS    9250         // [20]
#define OFF_CNT_T     9270         // [10]
#define OFF_CNT_S     9280         // [20]
#define WS_END        9300
#define ZERO_BEGIN    OFF_SUMT     // nt/ns fully overwritten; zero the rest

#define KC          32             // K slab per LDS stage
#define LDS_STRIDE  36             // 32 + pad: 144B rows -> 16B aligned, conflict-free b64

// ---------------- zero accumulators ----------------
__global__ void zero_ws_kernel(float* ws) {
    for (int i = ZERO_BEGIN + threadIdx.x; i < WS_END; i += blockDim.x)
        ws[i] = 0.0f;
}

// ---------------- row norms + class/group counts ----------------
// grid.x = N/8 (wave per row), grid.y = 2 (0=teacher,1=student)
__global__ __launch_bounds__(256) void rownorm_kernel(
    const float* __restrict__ ft, const float* __restrict__ fs,
    const int* __restrict__ labels, const int* __restrict__ groups,
    float* __restrict__ ws) {
    const int wave = threadIdx.x >> 5;
    const int lane = threadIdx.x & 31;
    const int row  = blockIdx.x * 8 + wave;
    const bool isT = (blockIdx.y == 0);
    const float* f = isT ? ft : fs;
    const float4* fr = (const float4*)(f + (size_t)row * DIM);
    float acc = 0.0f;
#pragma unroll
    for (int k = 0; k < 4; ++k) {            // 512 f32 = 128 float4 = 32 lanes * 4
        float4 v = fr[lane + 32 * k];
        acc += v.x * v.x + v.y * v.y + v.z * v.z + v.w * v.w;
    }
#pragma unroll
    for (int off = 16; off > 0; off >>= 1)   // wave32 reduction
        acc += __shfl_xor(acc, off, 32);
    if (lane == 0) {
        ws[(isT ? OFF_NT : OFF_NS) + row] = acc;
        atomicAdd(ws + (isT ? OFF_SUMNT : OFF_SUMNS), acc);
        if (isT) {                            // count each sample once
            const int lab = labels[row], grp = groups[row];
            atomicAdd(ws + OFF_CNT_T + lab, 1.0f);
            atomicAdd(ws + OFF_CNT_S + lab * NUM_GROUPS + grp, 1.0f);
        }
    }
}

// ---------------- column sums (for analytic sigma) ----------------
// grid: x = D/256, y = 16 row-chunks, z = 2 tensors
__global__ __launch_bounds__(256) void colsum_kernel(
    const float* __restrict__ ft, const float* __restrict__ fs, float* __restrict__ ws) {
    const int col = blockIdx.x * 256 + threadIdx.x;
    const float* f = (blockIdx.z == 0) ? ft : fs;
    const int r0 = blockIdx.y * (N_SAMPLES / 16);
    float acc = 0.0f;
    for (int i = 0; i < N_SAMPLES / 16; ++i)
        acc += f[(size_t)(r0 + i) * DIM + col];
    atomicAdd(ws + ((blockIdx.z == 0) ? OFF_SUMT : OFF_SUMS) + col, acc);
}

// ---------------- sigma / scale (one wave) ----------------
__global__ void scale_kernel(float* ws) {
    const int lane = threadIdx.x & 31;
    float dot = 0.0f;
    for (int k = lane; k < DIM; k += 32)
        dot += ws[OFF_SUMT + k] * ws[OFF_SUMS + k];
#pragma unroll
    for (int off = 16; off > 0; off >>= 1)
        dot += __shfl_xor(dot, off, 32);
    if (lane == 0) {
        const float n = (float)N_SAMPLES;
        // sum_ij d_ij = N*(sum||t||^2 + sum||s||^2) - 2*(sum t)·(sum s)
        const float sigma = (n * (ws[OFF_SUMNT] + ws[OFF_SUMNS]) - 2.0f * dot) / (n * n);
        ws[OFF_INVSCALE] = 1.0f / (2.0f * SIGMA_BASE_F * sigma);
    }
}

// ---------------- fused GEMM + exp + binning ----------------
// blockIdx.z: 0=TT (teacher x teacher), 1=SS (student x student), 2=TS (teacher x student)
// block = 256 threads = 8 waves, macro tile 64 rows x 32 cols, wave tile 16x16.
__global__ __launch_bounds__(256) void mmd_tiles_kernel(
    const float* __restrict__ ft, const float* __restrict__ fs,
    const int* __restrict__ labels, const int* __restrict__ groups,
    float* __restrict__ ws) {
    __shared__ float Xs[64 * LDS_STRIDE];
    __shared__ float Ys[32 * LDS_STRIDE];
    __shared__ float bins[CG];
    __shared__ float nxs[64], nys[32];
    __shared__ int   labx[64], grpx[64], laby[32], grpy[32];

    const int type = blockIdx.z;
    const float* X     = (type == 1) ? fs : ft;
    const float* Y     = (type == 0) ? ft : fs;
    const float* nxArr = ws + ((type == 1) ? OFF_NS : OFF_NT);
    const float* nyArr = ws + ((type == 0) ? OFF_NT : OFF_NS);

    const int tid  = threadIdx.x;
    const int lane = tid & 31;
    const int w    = tid >> 5;
    const int rowBase = blockIdx.y * 64;
    const int colBase = blockIdx.x * 32;

    // stage per-row / per-col metadata + zero bins (covered by first barrier)
    if (tid < 64) {
        const int i = rowBase + tid;
        nxs[tid] = nxArr[i]; labx[tid] = labels[i]; grpx[tid] = groups[i];
    } else if (tid < 96) {
        const int j = colBase + (tid - 64);
        nys[tid - 64] = nyArr[j]; laby[tid - 64] = labels[j]; grpy[tid - 64] = groups[j];
    } else if (tid < 96 + CG) {
        bins[tid - 96] = 0.0f;
    }
    const float inv_scale = ws[OFF_INVSCALE];

    // WMMA fragment addressing (ISA 16x16x4 f32 layout):
    //  A: lanes 0-15 hold M=lane, K={0,1}; lanes 16-31 hold M=lane-16, K={2,3}
    //  B: lanes 0-15 hold N=lane, K={0,1}; lanes 16-31 hold N=lane-16, K={2,3}
    const int hi = lane >> 4;
    const int lo = lane & 15;
    const int mrow = (w >> 1) * 16 + lo;   // local X row for A fragment
    const int nrow = (w & 1) * 16 + lo;    // local Y row for B fragment

    v8f c = {};
    for (int k0 = 0; k0 < DIM; k0 += KC) {
        __syncthreads();
        {   // stage X tile: 64 x KC   (each thread: two float4)
            const int r  = tid >> 3;          // 0..31
            const int cq = (tid & 7) * 4;     // 0..28
            *(float4*)&Xs[r * LDS_STRIDE + cq] =
                *(const float4*)&X[(size_t)(rowBase + r) * DIM + k0 + cq];
            *(float4*)&Xs[(r + 32) * LDS_STRIDE + cq] =
                *(const float4*)&X[(size_t)(rowBase + 32 + r) * DIM + k0 + cq];
            // stage Y tile: 32 x KC (one float4)
            *(float4*)&Ys[r * LDS_STRIDE + cq] =
                *(const float4*)&Y[(size_t)(colBase + r) * DIM + k0 + cq];
        }
        __syncthreads();
#pragma unroll
        for (int kk = 0; kk < KC; kk += 4) {
            v2f a = *(const v2f*)&Xs[mrow * LDS_STRIDE + kk + 2 * hi];
            v2f b = *(const v2f*)&Ys[nrow * LDS_STRIDE + kk + 2 * hi];
            c = __builtin_amdgcn_wmma_f32_16x16x4_f32(
                    false, a, false, b, (short)0, c, false, false);
        }
    }

    // epilogue: d = ||x||^2 + ||y||^2 - 2*G ; k = exp(-d*inv_scale); bin it.
    const int jj   = (w & 1) * 16 + lo;     // local col
    const float nyj = nys[jj];
    const int labj = laby[jj];
    const int grpj = grpy[jj];
#pragma unroll
    for (int r = 0; r < 8; ++r) {
        const int ii = (w >> 1) * 16 + r + 8 * hi;   // local row (C layout: M = r + 8*hi)
        float d = nxs[ii] + nyj - 2.0f * c[r];
        d = fmaxf(d, EPS_F);
        const float kv = __expf(-d * inv_scale);
        const int labi = labx[ii];
        bool match;
        int bin;
        if (type == 0)      { match = (labi == labj);                         bin = labi; }
        else if (type == 1) { match = (labi == labj) && (grpx[ii] == grpj);   bin = labi * NUM_GROUPS + grpj; }
        else                { match = (labi == labj);                         bin = labi * NUM_GROUPS + grpj; }
        if (match) atomicAdd(&bins[bin], kv);        // ds_add_f32, ~10% hit rate
    }
    __syncthreads();
    if (tid < CG) {
        const float v = bins[tid];
        if (v != 0.0f) {
            float* acc = ws + (type == 0 ? OFF_ACC_TT : (type == 1 ? OFF_ACC_SS : OFF_ACC_TS));
            atomicAdd(acc + tid, v);
        }
    }
}

// ---------------- finalize loss ----------------
__global__ void finalize_kernel(const float* __restrict__ ws, float* __restrict__ out) {
    if (threadIdx.x == 0) {
        float loss = 0.0f;
        for (int cc = 0; cc < NUM_CLASSES; ++cc) {
            const float nt  = ws[OFF_CNT_T + cc];
            const float snt = fmaxf(nt, 1.0f);
            const float mTT = ws[OFF_ACC_TT + cc] / (snt * snt);
            for (int g = 0; g < NUM_GROUPS; ++g) {
                const float nsv = ws[OFF_CNT_S + cc * NUM_GROUPS + g];
                const float sns = fmaxf(nsv, 1.0f);
                const float mSS = ws[OFF_ACC_SS + cc * NUM_GROUPS + g] / (sns * sns);
                const float mTS = ws[OFF_ACC_TS + cc * NUM_GROUPS + g] / (snt * sns);
                if (nt > 0.0f && nsv > 0.0f)
                    loss += mTT + mSS - 2.0f * mTS;
            }
        }
        out[0] = 0.5f * W_M_F * loss;
    }
}

extern "C" void kernel_launch(void* const* d_in, const int* in_sizes, int n_in,
                              void* d_out, int out_size, void* d_ws, size_t ws_size,
                              hipStream_t stream) {
    const float* f_s    = (const float*)d_in[0];
    const float* f_t    = (const float*)d_in[1];
    const int*   groups = (const int*)d_in[2];
    const int*   labels = (const int*)d_in[3];
    float* ws  = (float*)d_ws;
    float* out = (float*)d_out;

    zero_ws_kernel<<<1, 256, 0, stream>>>(ws);
    rownorm_kernel<<<dim3(N_SAMPLES / 8, 2), 256, 0, stream>>>(f_t, f_s, labels, groups, ws);
    colsum_kernel<<<dim3(DIM / 256, 16, 2), 256, 0, stream>>>(f_t, f_s, ws);
    scale_kernel<<<1, 32, 0, stream>>>(ws);
    mmd_tiles_kernel<<<dim3(N_SAMPLES / 32, N_SAMPLES / 64, 3), 256, 0, stream>>>(
        f_t, f_s, labels, groups, ws);
    finalize_kernel<<<1, 32, 0, stream>>>(ws, out);
}